// LPmodel_9268539425203
// MI455X (gfx1250) — compile-verified
//
#include <hip/hip_runtime.h>
#include <hip/hip_bf16.h>

typedef float v2f __attribute__((ext_vector_type(2)));
typedef float v8f __attribute__((ext_vector_type(8)));
typedef int   v4i __attribute__((vector_size(16)));

#ifndef __has_builtin
#define __has_builtin(x) 0
#endif

#if __has_builtin(__builtin_amdgcn_global_load_async_to_lds_b128)
#define HAVE_ASYNC_LDS 1
typedef __attribute__((address_space(1))) v4i* rgcn_gv4i_p;  // global int4*
typedef __attribute__((address_space(3))) v4i* rgcn_lv4i_p;  // LDS int4*
#else
#define HAVE_ASYNC_LDS 0
#endif

__device__ __forceinline__ void rgcn_wait_async0() {
#if __has_builtin(__builtin_amdgcn_s_wait_asynccnt)
  __builtin_amdgcn_s_wait_asynccnt(0);
#elif defined(__gfx1250__)
  asm volatile("s_wait_asynccnt 0x0" ::: "memory");
#endif
}

// ---------------------------------------------------------------- utilities
__global__ void rgcn_fill(float* __restrict__ p, size_t n, float v) {
  size_t i = (size_t)blockIdx.x * blockDim.x + threadIdx.x;
  size_t stride = (size_t)gridDim.x * blockDim.x;
  for (; i < n; i += stride) p[i] = v;
}

// deg_out[r][src], deg_in[r][dst] counting over all R*E edges
__global__ void rgcn_degree(const int* __restrict__ src, const int* __restrict__ dst,
                            float* __restrict__ deg_out, float* __restrict__ deg_in,
                            int total, int Nn, int Ee) {
  int idx = blockIdx.x * blockDim.x + threadIdx.x;
  if (idx >= total) return;
  int r = idx / Ee;
  atomicAdd(deg_out + (size_t)r * Nn + src[idx], 1.0f);
  atomicAdd(deg_in  + (size_t)r * Nn + dst[idx], 1.0f);
}

// d[i] = 1/sqrt(max(d[i], 1))
__global__ void rgcn_invsqrt(float* __restrict__ d, int n) {
  int i = blockIdx.x * blockDim.x + threadIdx.x;
  if (i >= n) return;
  float v = d[i];
  v = v < 1.0f ? 1.0f : v;
  d[i] = 1.0f / sqrtf(v);
}

// ------------------------------------------------- fp32 WMMA scaled GEMM
// Y[nrows, TN*16] = (H[nrows, FIN] * scale[row]) @ W[FIN, TN*16]
//
// Block = 256 threads = 8 waves. W is staged once per block into LDS
// (async global->LDS when available), rows padded by 4 dwords so the two
// K-rows read by lane halves hit disjoint LDS banks. Each wave owns one
// 16-row M-tile and computes the full 16 x Fout strip with TN accumulators,
// so each A fragment (V_WMMA_F32_16X16X4_F32 layout) feeds TN WMMAs.
//
// A-frag (16x4): lane m = lane&15; VGPR0/1 hold K = (lane>>4)*2 + {0,1}.
// B-frag (4x16): lane n = lane&15; VGPR0/1 hold same K pair.
// D (16x16): VGPR i holds row M = (lane>>4)*8 + i, col N = lane&15.
template <int TN, int FIN>
__global__ void __launch_bounds__(256)
rgcn_gemm_wmma(const float* __restrict__ H, const float* __restrict__ scale,
               const float* __restrict__ W, float* __restrict__ Y, int nrows) {
  constexpr int Fout = TN * 16;
  constexpr int LDW  = Fout + 4;            // row padding kills bank conflicts
  __shared__ float ldsW[FIN * LDW];

  // ---- stage W[FIN][Fout] into LDS (all threads, every block) ----
  constexpr int chunksPerRow = Fout / 4;    // float4 chunks
  constexpr int totalChunks  = FIN * chunksPerRow;
  for (int c = threadIdx.x; c < totalChunks; c += blockDim.x) {
    const int row  = c / chunksPerRow;
    const int col4 = (c - row * chunksPerRow) * 4;
    const float* g = W + (size_t)row * Fout + col4;
    float* l = &ldsW[row * LDW + col4];
#if HAVE_ASYNC_LDS
    __builtin_amdgcn_global_load_async_to_lds_b128(
        (rgcn_gv4i_p)(void*)g, (rgcn_lv4i_p)(void*)l, 0, 0);
#else
    *(float4*)l = *(const float4*)g;
#endif
  }
#if HAVE_ASYNC_LDS
  rgcn_wait_async0();
#endif
  __syncthreads();

  const int lane = threadIdx.x & 31;
  const int wave = threadIdx.x >> 5;
  const int tm = blockIdx.x * (blockDim.x >> 5) + wave;
  if (tm >= (nrows >> 4)) return;           // wave-uniform: EXEC stays all-1s

  const int lmod = lane & 15;
  const int lhi  = lane >> 4;               // 0 or 1
  const int kh   = lhi * 2;                 // K sub-offset (0 or 2)
  const int m    = tm * 16 + lmod;          // A row handled by this lane
  const float s  = scale[m];
  const float* __restrict__ arow = H + (size_t)m * FIN;

  v8f acc[TN];
#pragma unroll
  for (int t = 0; t < TN; ++t) acc[t] = {};

  for (int k = 0; k < FIN; k += 4) {
    v2f a;
    a.x = arow[k + kh]     * s;             // consecutive -> one b64 load
    a.y = arow[k + kh + 1] * s;
    const float* b0 = &ldsW[(k + kh) * LDW + lmod];
    const float* b1 = b0 + LDW;
#pragma unroll
    for (int t = 0; t < TN; ++t) {          // one A-frag feeds TN WMMAs
      v2f b;
      b.x = b0[t * 16];
      b.y = b1[t * 16];
      acc[t] = __builtin_amdgcn_wmma_f32_16x16x4_f32(
          /*neg_a=*/false, a, /*neg_b=*/false, b,
          /*c_mod=*/(short)0, acc[t], /*reuse_a=*/false, /*reuse_b=*/false);
    }
  }

  const size_t rowBase = (size_t)(tm * 16 + lhi * 8);
#pragma unroll
  for (int t = 0; t < TN; ++t) {
    const size_t base = rowBase * Fout + (size_t)(t * 16 + lmod);
#pragma unroll
    for (int i = 0; i < 8; ++i)
      Y[base + (size_t)i * Fout] = acc[t][i];
  }
}

// -------------------------------------------- edge gather + atomic scatter
// agg[dst[e]][f] += x[src[e]][f]   (agg buffers are L2-resident: 51 MB < 192 MB)
__global__ void rgcn_scatter(const float* __restrict__ x, const int* __restrict__ src,
                             const int* __restrict__ dst, float* __restrict__ agg,
                             int Ee, int F) {
  long long idx = (long long)blockIdx.x * blockDim.x + threadIdx.x;
  long long total = (long long)Ee * F;
  if (idx >= total) return;
  int e = (int)(idx / F);
  int f = (int)(idx - (long long)e * F);
  int sn = src[e], dn = dst[e];
  atomicAdd(agg + (size_t)dn * F + f, x[(size_t)sn * F + f]);
}

// hsum += agg * inv_in[node] + bias[f]   (accumulating the per-relation term)
__global__ void rgcn_accum(float* __restrict__ hsum, const float* __restrict__ agg,
                           const float* __restrict__ inv_in, const float* __restrict__ bias,
                           int Nn, int F) {
  long long idx = (long long)blockIdx.x * blockDim.x + threadIdx.x;
  long long total = (long long)Nn * F;
  if (idx >= total) return;
  int node = (int)(idx / F);
  int f = (int)(idx - (long long)node * F);
  hsum[idx] += agg[idx] * inv_in[node] + bias[f];
}

// h = (optionally relu)(h * scale)
__global__ void rgcn_finalize(float* __restrict__ h, size_t n, float scale, int do_relu) {
  size_t i = (size_t)blockIdx.x * blockDim.x + threadIdx.x;
  if (i >= n) return;
  float v = h[i] * scale;
  if (do_relu && v < 0.0f) v = 0.0f;
  h[i] = v;
}

// out[0:E] = dot(h[src0[e]], h[dst0[e]]);  out[E:2E] = dot(h[nsrc[e]], h[ndst[e]])
__global__ void rgcn_score(const float* __restrict__ h,
                           const int* __restrict__ src0, const int* __restrict__ dst0,
                           const int* __restrict__ nsrc, const int* __restrict__ ndst,
                           float* __restrict__ out, int Ee, int F) {
  long long idx = (long long)blockIdx.x * blockDim.x + threadIdx.x;
  if (idx >= 2LL * Ee) return;
  int e = (int)(idx % Ee);
  bool neg = idx >= Ee;
  int a = neg ? nsrc[e] : src0[e];
  int b = neg ? ndst[e] : dst0[e];
  const float4* pa = (const float4*)(h + (size_t)a * F);
  const float4* pb = (const float4*)(h + (size_t)b * F);
  float acc = 0.0f;
#pragma unroll 4
  for (int i = 0; i < F / 4; ++i) {
    float4 x = pa[i], y = pb[i];
    acc += x.x * y.x + x.y * y.y + x.z * y.z + x.w * y.w;
  }
  out[idx] = acc;
}

// ---------------------------------------------------------------- launcher
extern "C" void kernel_launch(void* const* d_in, const int* in_sizes, int n_in,
                              void* d_out, int out_size, void* d_ws, size_t ws_size,
                              hipStream_t stream) {
  (void)in_sizes; (void)n_in; (void)out_size; (void)ws_size;

  const float* feature = (const float*)d_in[0];   // [N, 256]
  const float* W1      = (const float*)d_in[1];   // [R, 256, 128]
  const float* b1      = (const float*)d_in[2];   // [R, 128]
  const float* W2      = (const float*)d_in[3];   // [R, 128, 64]
  const float* b2      = (const float*)d_in[4];   // [R, 64]
  const int*   src     = (const int*)d_in[5];     // [R, E]
  const int*   dst     = (const int*)d_in[6];     // [R, E]
  const int*   nsrc    = (const int*)d_in[7];     // [E]
  const int*   ndst    = (const int*)d_in[8];     // [E]
  float* out = (float*)d_out;                     // [2, E]

  const int Nn = 100000, R = 3, Ee = 500000;
  const int Fin = 256, Fhid = 128, Fo = 64;

  float* ws      = (float*)d_ws;
  float* deg_out = ws;                               // R*N
  float* deg_in  = deg_out + (size_t)R * Nn;         // R*N (contiguous with deg_out)
  float* xbuf    = deg_in  + (size_t)R * Nn;         // N*128
  float* aggbuf  = xbuf    + (size_t)Nn * Fhid;      // N*128
  float* h1      = aggbuf  + (size_t)Nn * Fhid;      // N*128
  float* h2      = h1      + (size_t)Nn * Fhid;      // N*64

  auto cdiv = [](long long a, long long b) { return (int)((a + b - 1) / b); };
  const int mtiles = Nn / 16;                        // 6250 (exact)
  const int gemmBlocks = cdiv(mtiles, 8);            // 8 waves/block, 1 M-tile/wave

  // zero accumulators (harness does not re-poison between replays)
  rgcn_fill<<<2048, 256, 0, stream>>>(deg_out, (size_t)2 * R * Nn, 0.0f);
  rgcn_fill<<<4096, 256, 0, stream>>>(h1, (size_t)Nn * Fhid, 0.0f);
  rgcn_fill<<<4096, 256, 0, stream>>>(h2, (size_t)Nn * Fo, 0.0f);

  // degrees -> 1/sqrt(clip(deg,1))
  rgcn_degree<<<cdiv((long long)R * Ee, 256), 256, 0, stream>>>(src, dst, deg_out, deg_in,
                                                                R * Ee, Nn, Ee);
  rgcn_invsqrt<<<cdiv((long long)2 * R * Nn, 256), 256, 0, stream>>>(deg_out, 2 * R * Nn);

  // ---- layer 1: relu(mean_r( D_in^-1/2 A_r (D_out^-1/2 X W1_r) + b1_r ))
  for (int r = 0; r < R; ++r) {
    rgcn_gemm_wmma<8, 256><<<gemmBlocks, 256, 0, stream>>>(
        feature, deg_out + (size_t)r * Nn, W1 + (size_t)r * Fin * Fhid, xbuf, Nn);
    rgcn_fill<<<4096, 256, 0, stream>>>(aggbuf, (size_t)Nn * Fhid, 0.0f);
    rgcn_scatter<<<cdiv((long long)Ee * Fhid, 256), 256, 0, stream>>>(
        xbuf, src + (size_t)r * Ee, dst + (size_t)r * Ee, aggbuf, Ee, Fhid);
    rgcn_accum<<<cdiv((long long)Nn * Fhid, 256), 256, 0, stream>>>(
        h1, aggbuf, deg_in + (size_t)r * Nn, b1 + (size_t)r * Fhid, Nn, Fhid);
  }
  rgcn_finalize<<<cdiv((long long)Nn * Fhid, 256), 256, 0, stream>>>(
      h1, (size_t)Nn * Fhid, 1.0f / 3.0f, /*relu=*/1);

  // ---- layer 2 (no relu)
  for (int r = 0; r < R; ++r) {
    rgcn_gemm_wmma<4, 128><<<gemmBlocks, 256, 0, stream>>>(
        h1, deg_out + (size_t)r * Nn, W2 + (size_t)r * Fhid * Fo, xbuf, Nn);
    rgcn_fill<<<4096, 256, 0, stream>>>(aggbuf, (size_t)Nn * Fo, 0.0f);
    rgcn_scatter<<<cdiv((long long)Ee * Fo, 256), 256, 0, stream>>>(
        xbuf, src + (size_t)r * Ee, dst + (size_t)r * Ee, aggbuf, Ee, Fo);
    rgcn_accum<<<cdiv((long long)Nn * Fo, 256), 256, 0, stream>>>(
        h2, aggbuf, deg_in + (size_t)r * Nn, b2 + (size_t)r * Fo, Nn, Fo);
  }
  rgcn_finalize<<<cdiv((long long)Nn * Fo, 256), 256, 0, stream>>>(
      h2, (size_t)Nn * Fo, 1.0f / 3.0f, /*relu=*/0);

  // ---- scores: pos uses relation 0 edges, neg uses neg_src/neg_dst
  rgcn_score<<<cdiv(2LL * Ee, 256), 256, 0, stream>>>(h2, src, dst, nsrc, ndst, out, Ee, Fo);
}